// RNNEncoder_13958643712428
// MI455X (gfx1250) — compile-verified
//
#include <hip/hip_runtime.h>

// LSTM encoder: H=4, L=5, IN=1, B=1024, T=2048 -> out [1024,4] (final h of top layer)
//
// V_WMMA_F32_16X16X4_F32: A = gate-permuted, sigmoid-prescaled W (16x4),
// B = h^T (4x16 batch), C = gates[gate_row, batch]. Gate rows are permuted so
// each lane's own 8 C-registers hold exactly the i/f/g/o values for the 2
// hidden units that lane owns -> no cross-lane traffic. The 0.5 factor of the
// tanh-based sigmoid is folded into the i/f/o weight rows and biases, so the
// loop needs no pre-scale muls. h state lives directly in the WMMA B-operand
// layout. All 5 layers fused in registers; 2 batch tiles per wave for ILP;
// x loads software-pipelined one group (4 timesteps) ahead.

typedef float v2f __attribute__((ext_vector_type(2)));
typedef float v4f __attribute__((ext_vector_type(4)));
typedef float v8f __attribute__((ext_vector_type(8)));

#define HID 4
#define NL  5
#define BATCH 1024
#define SEQT  2048

// Permuted gate-row -> original gate index in the 4H weight row dimension.
// rows 0-7: i0,i1,f0,f1,g0,g1,o0,o1   rows 8-15: i2,i3,f2,f3,g2,g3,o2,o3
__device__ __forceinline__ int gate_orig(int m) {
    return (((m >> 1) & 3) << 2) + ((m >> 3) << 1) + (m & 1);
}
// 0.5 for sigmoid rows (i,f,o), 1.0 for cell-gate rows (g = group 2)
__device__ __forceinline__ float gate_scale(int m) {
    return (((m >> 1) & 3) == 2) ? 1.0f : 0.5f;
}

__device__ __forceinline__ float fast_tanh(float x) {
#if __has_builtin(__builtin_amdgcn_tanhf)
    return __builtin_amdgcn_tanhf(x);           // v_tanh_f32 (gfx1250 TRANS op)
#else
    float e = __builtin_amdgcn_exp2f(-2.885390082f * x);
    return __builtin_fmaf(2.0f, __builtin_amdgcn_rcpf(1.0f + e), -1.0f);
#endif
}
// input y is already 0.5 * preactivation (scale folded into W,b)
__device__ __forceinline__ float sig_half(float y) {
    return __builtin_fmaf(fast_tanh(y), 0.5f, 0.5f);
}

__global__ __launch_bounds__(32)
void lstm5_wmma_kernel(const float* __restrict__ x,        // [B, T, 1]
                       const float* __restrict__ w_ih0,    // [16, 1]
                       const float* __restrict__ w_ihr,    // [4, 16, 4]
                       const float* __restrict__ w_hh,     // [5, 16, 4]
                       const float* __restrict__ b_ih,     // [5, 16]
                       const float* __restrict__ b_hh,     // [5, 16]
                       float* __restrict__ out)            // [B, 4]
{
    const int   lane = threadIdx.x;         // wave32, one wave per block
    const int   ml   = lane & 15;           // A row M / B column N
    const bool  hi   = lane >= 16;          // half selects K=2,3 (vs K=0,1)
    const int   k0   = hi ? 2 : 0;
    const int   arow = gate_orig(ml);       // original gate for this lane's A row
    const float asc  = gate_scale(ml);      // sigmoid 0.5 folded into W rows

    // ---- preload weights into WMMA A-layout (lane = M, half/VGPR = K) ----
    v2f a_hh[NL], a_ih[NL];
    v8f cb[NL];                              // bias in (permuted, scaled) C layout
#pragma unroll
    for (int l = 0; l < NL; ++l) {
        a_hh[l].x = asc * w_hh[l * 64 + arow * 4 + k0];
        a_hh[l].y = asc * w_hh[l * 64 + arow * 4 + k0 + 1];
    }
    a_ih[0].x = hi ? 0.0f : asc * w_ih0[arow];   // K=0 column only; rest zero-pad
    a_ih[0].y = 0.0f;
#pragma unroll
    for (int l = 1; l < NL; ++l) {
        a_ih[l].x = asc * w_ihr[(l - 1) * 64 + arow * 4 + k0];
        a_ih[l].y = asc * w_ihr[(l - 1) * 64 + arow * 4 + k0 + 1];
    }
#pragma unroll
    for (int l = 0; l < NL; ++l) {
#pragma unroll
        for (int r = 0; r < 8; ++r) {
            int   m = r + (hi ? 8 : 0);
            int   g = gate_orig(m);
            cb[l][r] = gate_scale(m) * (b_ih[l * 16 + g] + b_hh[l * 16 + g]);
        }
    }

    // ---- two batch tiles of 16 per wave (independent chains for ILP) ----
    const int nb = blockIdx.x * 32;
    const int n0 = nb + ml;
    const int n1 = nb + 16 + ml;
    const v4f* xr0 = (const v4f*)(x + (size_t)n0 * SEQT);  // rows 8KB apart, 16B aligned
    const v4f* xr1 = (const v4f*)(x + (size_t)n1 * SEQT);

    v2f hB[2][NL];   // h in WMMA B-operand layout: lane=batch, half/reg = K pair
    v2f cs[2][NL];   // c for the 2 hidden units this lane owns (j = k0, k0+1)
#pragma unroll
    for (int u = 0; u < 2; ++u)
#pragma unroll
        for (int l = 0; l < NL; ++l) {
            hB[u][l] = (v2f){0.0f, 0.0f};
            cs[u][l] = (v2f){0.0f, 0.0f};
        }

    constexpr int NG = SEQT / 4;
    v4f cur0 = xr0[0];                       // software-pipelined x stream
    v4f cur1 = xr1[0];

    for (int tg = 0; tg < NG; ++tg) {
        const int tn = (tg + 1 < NG) ? (tg + 1) : tg;   // clamped prefetch index
        v4f nxt0 = xr0[tn];
        v4f nxt1 = xr1[tn];

#pragma unroll
        for (int q = 0; q < 4; ++q) {
            v2f bin[2];
            bin[0] = (v2f){hi ? 0.0f : cur0[q], 0.0f};   // B[0,n]=x, rest 0
            bin[1] = (v2f){hi ? 0.0f : cur1[q], 0.0f};

#pragma unroll
            for (int l = 0; l < NL; ++l) {
#pragma unroll
                for (int u = 0; u < 2; ++u) {
                    // gates = W_ih*input + bias, then += W_hh*h_{t-1}
                    v8f g1 = __builtin_amdgcn_wmma_f32_16x16x4_f32(
                        false, a_ih[l], false, bin[u], (short)0, cb[l], false, false);
                    v8f gg = __builtin_amdgcn_wmma_f32_16x16x4_f32(
                        false, a_hh[l], false, hB[u][l], (short)0, g1, false, false);

                    // permuted rows: gg = [Ia, Ib, Fa, Fb, Ga, Gb, Oa, Ob];
                    // i/f/o rows arrive pre-scaled by 0.5 for the sigmoid.
                    float tG0 = fast_tanh(gg[4]);
                    float tG1 = fast_tanh(gg[5]);
                    float c0  = sig_half(gg[2]) * cs[u][l].x + sig_half(gg[0]) * tG0;
                    float c1  = sig_half(gg[3]) * cs[u][l].y + sig_half(gg[1]) * tG1;
                    float h0  = sig_half(gg[6]) * fast_tanh(c0);
                    float h1  = sig_half(gg[7]) * fast_tanh(c1);

                    cs[u][l] = (v2f){c0, c1};
                    hB[u][l] = (v2f){h0, h1};   // already in B layout
                    bin[u]   = hB[u][l];        // feeds next layer's input WMMA
                }
            }
        }
        cur0 = nxt0;
        cur1 = nxt1;
    }

    // final hidden state of top layer: lane writes its 2 owned units
    out[n0 * HID + k0]     = hB[0][NL - 1].x;
    out[n0 * HID + k0 + 1] = hB[0][NL - 1].y;
    out[n1 * HID + k0]     = hB[1][NL - 1].x;
    out[n1 * HID + k0 + 1] = hB[1][NL - 1].y;
}

extern "C" void kernel_launch(void* const* d_in, const int* in_sizes, int n_in,
                              void* d_out, int out_size, void* d_ws, size_t ws_size,
                              hipStream_t stream) {
    (void)in_sizes; (void)n_in; (void)out_size; (void)d_ws; (void)ws_size;
    const float* x     = (const float*)d_in[0];
    const float* w_ih0 = (const float*)d_in[1];
    const float* w_ihr = (const float*)d_in[2];
    const float* w_hh  = (const float*)d_in[3];
    const float* b_ih  = (const float*)d_in[4];
    const float* b_hh  = (const float*)d_in[5];
    float* out = (float*)d_out;

    // 32 blocks x 1 wave; each wave handles 32 batch rows (2 WMMA tiles)
    lstm5_wmma_kernel<<<BATCH / 32, 32, 0, stream>>>(x, w_ih0, w_ihr, w_hh,
                                                     b_ih, b_hh, out);
}